// TaskGNN_64424509440459
// MI455X (gfx1250) — compile-verified
//
#include <hip/hip_runtime.h>
#include <hip/hip_bf16.h>
#include <math.h>

// ---------------------------------------------------------------------------
// GAT (2 layers + fc) for MI455X / gfx1250.
//   N=50000 nodes, E=800000 edges (+N self loops), IN=128, H=4, C=32, OUT=64.
// GEMMs use V_WMMA_F32_16X16X4_F32 (exact fp32 matrix pipe); edge softmax /
// aggregation use coalesced gathers + GLOBAL_ATOMIC_{MAX_U32, ADD_F32}.
// ---------------------------------------------------------------------------

#define N_NODES   50000
#define N_EDGES   800000
#define ET_EDGES  (N_EDGES + N_NODES)   // with self loops
#define IN_DIM    128
#define HID       32
#define HEADS     4
#define OUT_DIM   64
#define NEG_SLOPE 0.2f

typedef __attribute__((ext_vector_type(2))) float v2f;
typedef __attribute__((ext_vector_type(8))) float v8f;

// ---- monotone float<->uint encoding so we can use GLOBAL_ATOMIC_MAX_U32 ----
__device__ __forceinline__ unsigned f32_to_ord(float f) {
  unsigned u = __float_as_uint(f);
  return (u >> 31) ? ~u : (u | 0x80000000u);
}
__device__ __forceinline__ float ord_to_f32(unsigned u) {
  u = (u >> 31) ? (u & 0x7fffffffu) : ~u;
  return __uint_as_float(u);
}
#define ORD_NEG_INF 0x007FFFFFu   // f32_to_ord(-inf)

// ---------------------------------------------------------------------------
// GEMM  C[M x Nc] = A[M x K] * B[K x Nc] (+ bias), all row-major fp32.
// One wave computes one 16x16 tile via v_wmma_f32_16x16x4_f32; waves of a
// block tile the N dimension (blockDim.x = 32 * Nc/16); A-tile staged in LDS.
//
// fp32 WMMA fragment layout (wave32):
//   A (16x4):  lanes 0-15 -> M=lane, VGPR0/1 = K0/K1; lanes 16-31 -> K2/K3
//   B (4x16):  lanes 0-15 -> N=lane, VGPR0/1 = K0/K1; lanes 16-31 -> K2/K3
//   C (16x16): VGPR j: lanes 0-15 -> M=j, lanes 16-31 -> M=j+8, N=lane%16
// ---------------------------------------------------------------------------
__global__ void gemm16_wmma_f32(const float* __restrict__ A,
                                const float* __restrict__ B,
                                const float* __restrict__ bias,
                                float* __restrict__ C,
                                int M, int K, int Nc) {
  extern __shared__ float sA[];              // 16 x K
  const int wave = threadIdx.x >> 5;
  const int lane = threadIdx.x & 31;
  const int row0 = blockIdx.x * 16;
  const int col0 = wave * 16;

  for (int i = threadIdx.x; i < 16 * K; i += blockDim.x) {
    int r = i / K, c = i - r * K;
    int gr = row0 + r;
    sA[i] = (gr < M) ? A[(size_t)gr * K + c] : 0.0f;
  }
  __syncthreads();

  const int lrow  = lane & 15;
  const int khalf = (lane >> 4) << 1;        // 0 or 2
  v8f acc = {0.f, 0.f, 0.f, 0.f, 0.f, 0.f, 0.f, 0.f};

  for (int k = 0; k < K; k += 4) {
    v2f a, b;
    a.x = sA[lrow * K + k + khalf];
    a.y = sA[lrow * K + k + khalf + 1];
    b.x = B[(size_t)(k + khalf)     * Nc + col0 + lrow];
    b.y = B[(size_t)(k + khalf + 1) * Nc + col0 + lrow];
    acc = __builtin_amdgcn_wmma_f32_16x16x4_f32(
        /*neg_a=*/false, a, /*neg_b=*/false, b,
        /*c_mod=*/(short)0, acc, /*reuse_a=*/false, /*reuse_b=*/false);
  }

  float bv = bias ? bias[col0 + lrow] : 0.0f;
#pragma unroll
  for (int j = 0; j < 8; ++j) {
    int r = row0 + ((lane < 16) ? j : (j + 8));
    if (r < M) C[(size_t)r * Nc + col0 + lrow] = acc[j] + bv;
  }
}

// ---------------------------------------------------------------------------
// alpha_s[n,h] = sum_c h[n,h,c]*a_src[h,c] ; alpha_d likewise
// ---------------------------------------------------------------------------
__global__ void attn_coef_kernel(const float* __restrict__ h,
                                 const float* __restrict__ a_src,
                                 const float* __restrict__ a_dst,
                                 float* __restrict__ as_, float* __restrict__ ad_,
                                 int n, int H, int C) {
  int t = blockIdx.x * blockDim.x + threadIdx.x;
  if (t >= n * H) return;
  int node = t / H, hh = t - node * H;
  const float* hp = h + (size_t)node * H * C + (size_t)hh * C;
  const float* ap = a_src + hh * C;
  const float* bp = a_dst + hh * C;
  float ss = 0.f, sd = 0.f;
  for (int c = 0; c < C; ++c) { float v = hp[c]; ss += v * ap[c]; sd += v * bp[c]; }
  as_[t] = ss; ad_[t] = sd;
}

__global__ void fill_u32_kernel(unsigned* p, unsigned v, int n) {
  int t = blockIdx.x * blockDim.x + threadIdx.x;
  if (t < n) p[t] = v;
}
__global__ void fill_f32_kernel(float* p, float v, int n) {
  int t = blockIdx.x * blockDim.x + threadIdx.x;
  if (t < n) p[t] = v;
}

__device__ __forceinline__ void edge_endpoints(const long long* __restrict__ ei,
                                               int t, int& s, int& d) {
  if (t < N_EDGES) { s = (int)ei[t]; d = (int)ei[N_EDGES + t]; }
  else             { s = d = t - N_EDGES; }                     // self loop
}

// segment max of leaky_relu(as[src]+ad[dst]) into ordered-uint mo[dst,h]
__global__ void edge_max_kernel(const long long* __restrict__ ei,
                                const float* __restrict__ as_,
                                const float* __restrict__ ad_,
                                unsigned* __restrict__ mo, int H) {
  int t = blockIdx.x * blockDim.x + threadIdx.x;
  if (t >= ET_EDGES) return;
  int s, d; edge_endpoints(ei, t, s, d);
  for (int h = 0; h < H; ++h) {
    float e = as_[s * H + h] + ad_[d * H + h];
    e = (e > 0.f) ? e : NEG_SLOPE * e;
    atomicMax(&mo[d * H + h], f32_to_ord(e));
  }
}

// e = exp(lrelu(..) - max[dst]); store per edge; segment-sum into ssum[dst,h]
__global__ void edge_exp_kernel(const long long* __restrict__ ei,
                                const float* __restrict__ as_,
                                const float* __restrict__ ad_,
                                const unsigned* __restrict__ mo,
                                float* __restrict__ ebuf,
                                float* __restrict__ ssum, int H) {
  int t = blockIdx.x * blockDim.x + threadIdx.x;
  if (t >= ET_EDGES) return;
  int s, d; edge_endpoints(ei, t, s, d);
  for (int h = 0; h < H; ++h) {
    float e = as_[s * H + h] + ad_[d * H + h];
    e = (e > 0.f) ? e : NEG_SLOPE * e;
    float ex = expf(e - ord_to_f32(mo[d * H + h]));
    ebuf[(size_t)t * H + h] = ex;
    atomicAdd(&ssum[d * H + h], ex);
  }
}

// out[dst, c] += h[src, c] * ebuf[t,h]/(ssum[dst,h]+1e-16)
// one thread per (edge, channel): lanes cover consecutive channels of one
// edge -> coalesced gather + coalesced scatter atomics (L2-resident).
__global__ void edge_aggregate_kernel(const long long* __restrict__ ei,
                                      const float* __restrict__ hfeat,
                                      const float* __restrict__ ebuf,
                                      const float* __restrict__ ssum,
                                      float* __restrict__ out, int H, int C) {
  long long tid = (long long)blockIdx.x * blockDim.x + threadIdx.x;
  const int HC = H * C;
  if (tid >= (long long)ET_EDGES * HC) return;
  int t = (int)(tid / HC);
  int c = (int)(tid - (long long)t * HC);
  int h = c / C;
  int s, d; edge_endpoints(ei, t, s, d);
  float alpha = ebuf[(size_t)t * H + h] / (ssum[(size_t)d * H + h] + 1e-16f);
  atomicAdd(&out[(size_t)d * HC + c], hfeat[(size_t)s * HC + c] * alpha);
}

// v[n,c] = elu(v[n,c] + b[c])
__global__ void bias_elu_kernel(float* __restrict__ v, const float* __restrict__ b,
                                int n, int F) {
  int t = blockIdx.x * blockDim.x + threadIdx.x;
  if (t >= n * F) return;
  int c = t % F;
  float x = v[t] + b[c];
  v[t] = (x > 0.f) ? x : expm1f(x);
}

// ---------------------------------------------------------------------------
extern "C" void kernel_launch(void* const* d_in, const int* in_sizes, int n_in,
                              void* d_out, int out_size, void* d_ws, size_t ws_size,
                              hipStream_t stream) {
  (void)in_sizes; (void)n_in; (void)out_size; (void)ws_size;
  const float*     x      = (const float*)d_in[0];
  const long long* ei     = (const long long*)d_in[1];   // int64 [2, E]
  const float*     W1     = (const float*)d_in[2];
  const float*     a_src1 = (const float*)d_in[3];
  const float*     a_dst1 = (const float*)d_in[4];
  const float*     b1     = (const float*)d_in[5];
  const float*     W2     = (const float*)d_in[6];
  const float*     a_src2 = (const float*)d_in[7];
  const float*     a_dst2 = (const float*)d_in[8];
  const float*     b2     = (const float*)d_in[9];
  const float*     fc_w   = (const float*)d_in[10];
  const float*     fc_b   = (const float*)d_in[11];
  float*           out    = (float*)d_out;

  // ---- workspace carve-up (~68 MB) ----
  float* ws = (float*)d_ws;
  size_t off = 0;
  float* h1   = ws + off; off += (size_t)N_NODES * 128;   // layer-1 features
  float* act1 = ws + off; off += (size_t)N_NODES * 128;   // layer-1 output (elu)
  float* ebuf = ws + off; off += (size_t)ET_EDGES * HEADS;
  float* as_  = ws + off; off += (size_t)N_NODES * HEADS;
  float* ad_  = ws + off; off += (size_t)N_NODES * HEADS;
  unsigned* mo = (unsigned*)(ws + off); off += (size_t)N_NODES * HEADS;
  float* ssum = ws + off; off += (size_t)N_NODES * HEADS;
  // layer-2 reuses the dead h1 region
  float* h2   = h1;                                        // [N, 64]
  float* out2 = h1 + (size_t)N_NODES * 64;                 // [N, 64]

  const int rowTiles = (N_NODES + 15) / 16;                // 3125
  const int BLK = 256;
  const int gET  = (ET_EDGES + BLK - 1) / BLK;

  // ================= layer 1  (H=4, C=32, F=128) =================
  gemm16_wmma_f32<<<rowTiles, 8 * 32, 16 * IN_DIM * sizeof(float), stream>>>(
      x, W1, nullptr, h1, N_NODES, IN_DIM, HEADS * HID);

  attn_coef_kernel<<<(N_NODES * HEADS + BLK - 1) / BLK, BLK, 0, stream>>>(
      h1, a_src1, a_dst1, as_, ad_, N_NODES, HEADS, HID);

  fill_u32_kernel<<<(N_NODES * HEADS + BLK - 1) / BLK, BLK, 0, stream>>>(
      mo, ORD_NEG_INF, N_NODES * HEADS);
  fill_f32_kernel<<<(N_NODES * HEADS + BLK - 1) / BLK, BLK, 0, stream>>>(
      ssum, 0.f, N_NODES * HEADS);
  fill_f32_kernel<<<(N_NODES * 128 + BLK - 1) / BLK, BLK, 0, stream>>>(
      act1, 0.f, N_NODES * 128);

  edge_max_kernel<<<gET, BLK, 0, stream>>>(ei, as_, ad_, mo, HEADS);
  edge_exp_kernel<<<gET, BLK, 0, stream>>>(ei, as_, ad_, mo, ebuf, ssum, HEADS);
  {
    long long total = (long long)ET_EDGES * (HEADS * HID);
    edge_aggregate_kernel<<<(unsigned)((total + BLK - 1) / BLK), BLK, 0, stream>>>(
        ei, h1, ebuf, ssum, act1, HEADS, HID);
  }
  bias_elu_kernel<<<(N_NODES * 128 + BLK - 1) / BLK, BLK, 0, stream>>>(
      act1, b1, N_NODES, 128);

  // ================= layer 2  (H=1, C=64) =================
  gemm16_wmma_f32<<<rowTiles, 4 * 32, 16 * 128 * sizeof(float), stream>>>(
      act1, W2, nullptr, h2, N_NODES, 128, OUT_DIM);

  attn_coef_kernel<<<(N_NODES + BLK - 1) / BLK, BLK, 0, stream>>>(
      h2, a_src2, a_dst2, as_, ad_, N_NODES, 1, OUT_DIM);

  fill_u32_kernel<<<(N_NODES + BLK - 1) / BLK, BLK, 0, stream>>>(
      mo, ORD_NEG_INF, N_NODES);
  fill_f32_kernel<<<(N_NODES + BLK - 1) / BLK, BLK, 0, stream>>>(
      ssum, 0.f, N_NODES);
  fill_f32_kernel<<<(N_NODES * OUT_DIM + BLK - 1) / BLK, BLK, 0, stream>>>(
      out2, 0.f, N_NODES * OUT_DIM);

  edge_max_kernel<<<gET, BLK, 0, stream>>>(ei, as_, ad_, mo, 1);
  edge_exp_kernel<<<gET, BLK, 0, stream>>>(ei, as_, ad_, mo, ebuf, ssum, 1);
  {
    long long total = (long long)ET_EDGES * OUT_DIM;
    edge_aggregate_kernel<<<(unsigned)((total + BLK - 1) / BLK), BLK, 0, stream>>>(
        ei, h2, ebuf, ssum, out2, 1, OUT_DIM);
  }
  bias_elu_kernel<<<(N_NODES * OUT_DIM + BLK - 1) / BLK, BLK, 0, stream>>>(
      out2, b2, N_NODES, OUT_DIM);

  // ================= final fc: out = out2 @ fc_w + fc_b =================
  gemm16_wmma_f32<<<rowTiles, 4 * 32, 16 * OUT_DIM * sizeof(float), stream>>>(
      out2, fc_w, fc_b, out, N_NODES, OUT_DIM, OUT_DIM);
}